// CrossFrameAttention_13408887898646
// MI455X (gfx1250) — compile-verified
//
#include <hip/hip_runtime.h>

typedef __attribute__((ext_vector_type(2))) float v2f;
typedef __attribute__((ext_vector_type(8))) float v8f;

#define SAI_N 9
#define B_N   4
#define CK_N  64
#define CV_N  64
#define HW_N  1024              // 32*32
#define SHW_N (SAI_N * HW_N)    // 9216
#define RADIUS_F 0.5f
#define NEG_BIG (-1.0e30f)

#define NW 8                              // waves per block (split-K ways)
#define KEYS_PER_WAVE (SHW_N / NW)        // 1152 keys per wave (72 chunks)

// 8 waves per block; each wave owns the full 16-query tile but a disjoint
// 1152-key slice (flash-attention split-K), iterated as <=2 frame segments
// so the per-frame distance factor is loop-invariant in the chunk loop.
// Partial (m,l,O) merge through LDS at the end. fp32 WMMA throughout.
__global__ __launch_bounds__(32 * NW)
void cfa_fp32_wmma_splitk(const float* __restrict__ mem_keys,   // (SAI,B,CK,H,W)
                          const float* __restrict__ mem_vals,   // (SAI,B,CV,H,W)
                          const float* __restrict__ query,      // (B,CK,H,W)
                          const float* __restrict__ disparity,  // (B,1,H,W)
                          const int*   __restrict__ seq_index,  // (B,SAI,2)
                          float* __restrict__ out)              // (B,CV,H,W)
{
    __shared__ float pt[NW][16 * 16];       // per-wave P-tile re-layout buffer
    __shared__ float lm[NW][16];            // per-wave row maxima
    __shared__ float ll[NW][16];            // per-wave row sums
    __shared__ float lo[NW][16][CV_N];      // per-wave unnormalized O tiles

    const int tid  = threadIdx.x;
    const int w    = tid >> 5;             // wave id within block
    const int lane = tid & 31;
    const int b    = blockIdx.x >> 6;      // batch
    const int q0   = (blockIdx.x & 63) * 16;

    const int m    = lane & 15;            // A-layout row (and B/C/D column)
    const int kb   = (lane >> 4) * 2;      // A/B-layout k-pair base
    const int n    = lane & 15;            // B/C/D column index
    const int half = lane >> 4;            // 0: rows v, 1: rows v+8 (C/D layout)

    // ---- preload Q tile as 16 A-fragments (channels cc*4 .. cc*4+3) ----
    v2f qa[16];
    {
        const float* qbase = query + (size_t)b * CK_N * HW_N + q0 + m;
        #pragma unroll
        for (int cc = 0; cc < 16; ++cc) {
            const int c0 = cc * 4 + kb;
            qa[cc].x = qbase[(size_t)c0 * HW_N];
            qa[cc].y = qbase[(size_t)(c0 + 1) * HW_N];
        }
    }

    // ---- online-softmax running state (8 rows per lane-half).
    // lsum is a PER-LANE partial sum: the rescale factor is uniform across
    // each 16-lane half, so the cross-lane sum reduction is deferred to the
    // end (one reduction per kernel instead of one per chunk).
    float mrow[8], lsum[8];
    #pragma unroll
    for (int v = 0; v < 8; ++v) { mrow[v] = NEG_BIG; lsum[v] = 0.0f; }
    v8f oacc[4] = {};

    // ---- key slice [kstart, kend): at most 2 frame segments ----
    const int kstart = w * KEYS_PER_WAVE;
    const int kend   = kstart + KEYS_PER_WAVE;
    for (int s = kstart >> 10; s * HW_N < kend; ++s) {
        const int klo = (kstart > s * HW_N) ? kstart : s * HW_N;
        const int khi = (kend < (s + 1) * HW_N) ? kend : (s + 1) * HW_N;

        // frame distance factor (loop-invariant over the chunk loop)
        const float dx = (float)seq_index[(b * SAI_N + s) * 2 + 0] - 5.0f;
        const float dy = (float)seq_index[(b * SAI_N + s) * 2 + 1] - 5.0f;
        const float df = sqrtf(dx * dx + dy * dy);

        const float* kframe = mem_keys + (((size_t)s * B_N + b) * CK_N) * HW_N;
        const float* vframe = mem_vals + (((size_t)s * B_N + b) * CV_N) * HW_N;

        for (int k0 = klo; k0 < khi; k0 += 16) {
            const int p0 = k0 & (HW_N - 1);
            const float* kbase = kframe + p0;
            const float* vbase = vframe + p0;

            __builtin_prefetch((const void*)(kbase + 16), 0, 1); // global_prefetch_b8

            // ---- S-tile = Q x K-chunk : 16 fp32 WMMAs ----
            v8f sc = {};
            #pragma unroll
            for (int cc = 0; cc < 16; ++cc) {
                const int c0 = cc * 4 + kb;
                v2f bk;
                bk.x = kbase[(size_t)c0 * HW_N + n];
                bk.y = kbase[(size_t)(c0 + 1) * HW_N + n];
                sc = __builtin_amdgcn_wmma_f32_16x16x4_f32(
                    false, qa[cc], false, bk, (short)0, sc, false, false);
            }

            // ---- additive mask bias: per key column, same for all rows ----
            const float disp = disparity[(size_t)b * HW_N + p0 + n];
            const float bias = (fabsf(df * disp) > RADIUS_F) ? 0.0f : NEG_BIG;
            #pragma unroll
            for (int v = 0; v < 8; ++v) sc[v] += bias;

            // ---- online softmax: row max (cross-lane) + per-lane sums ----
            float pfrag[8];
            #pragma unroll
            for (int v = 0; v < 8; ++v) {
                float cm = sc[v];                   // row max over 16 lanes/half
                #pragma unroll
                for (int off = 1; off < 16; off <<= 1)
                    cm = fmaxf(cm, __shfl_xor(cm, off, 32));
                const float mn    = fmaxf(mrow[v], cm);
                const float scale = __expf(mrow[v] - mn);
                const float pv    = __expf(sc[v] - mn);
                lsum[v] = lsum[v] * scale + pv;     // per-lane partial sum
                mrow[v] = mn;
                #pragma unroll
                for (int t = 0; t < 4; ++t) oacc[t][v] *= scale;
                pfrag[v] = pv;
            }

            // ---- stage P through per-wave LDS: C/D -> A operand layout.
            // Same-wave DS ops are processed in order (ISA 7.3); only a
            // compiler scheduling fence is needed.
            #pragma unroll
            for (int v = 0; v < 8; ++v)
                pt[w][(v + 8 * half) * 16 + n] = pfrag[v];
            __builtin_amdgcn_wave_barrier();

            // ---- O += P x V-chunk : 4 tiles x 4 k-subchunks fp32 WMMAs ----
            #pragma unroll
            for (int j = 0; j < 4; ++j) {
                v2f pa;
                pa.x = pt[w][m * 16 + j * 4 + kb];
                pa.y = pt[w][m * 16 + j * 4 + kb + 1];
                #pragma unroll
                for (int t = 0; t < 4; ++t) {
                    const int cv0 = t * 16 + n;
                    v2f bv;
                    bv.x = vbase[(size_t)cv0 * HW_N + (j * 4 + kb)];
                    bv.y = vbase[(size_t)cv0 * HW_N + (j * 4 + kb + 1)];
                    oacc[t] = __builtin_amdgcn_wmma_f32_16x16x4_f32(
                        false, pa, false, bv, (short)0, oacc[t], false, false);
                }
            }
            __builtin_amdgcn_wave_barrier();  // pt reused next iteration
        }
    }

    // ---- finalize deferred row sums (single cross-lane reduction) ----
    float lrow[8];
    #pragma unroll
    for (int v = 0; v < 8; ++v) {
        float rs = lsum[v];
        #pragma unroll
        for (int off = 1; off < 16; off <<= 1)
            rs += __shfl_xor(rs, off, 32);
        lrow[v] = rs;
    }

    // ---- publish this wave's partial state to LDS ----
    if (n == 0) {                      // lanes 0 and 16: one writer per half
        #pragma unroll
        for (int v = 0; v < 8; ++v) {
            lm[w][v + 8 * half] = mrow[v];
            ll[w][v + 8 * half] = lrow[v];
        }
    }
    #pragma unroll
    for (int v = 0; v < 8; ++v) {
        const int row = v + 8 * half;
        #pragma unroll
        for (int t = 0; t < 4; ++t)
            lo[w][row][t * 16 + n] = oacc[t][v];
    }
    __syncthreads();

    // ---- cross-wave flash-attention merge: 1024 outputs, 4 per thread ----
    {
        const int row = tid >> 4;            // 0..15
        const int cv0 = (tid & 15) * 4;      // 0,4,..,60

        float mstar = NEG_BIG;
        #pragma unroll
        for (int w2 = 0; w2 < NW; ++w2) mstar = fmaxf(mstar, lm[w2][row]);

        float lstar = 0.0f;
        float acc[4] = {0.0f, 0.0f, 0.0f, 0.0f};
        #pragma unroll
        for (int w2 = 0; w2 < NW; ++w2) {
            const float f = __expf(lm[w2][row] - mstar);
            lstar += ll[w2][row] * f;
            #pragma unroll
            for (int j = 0; j < 4; ++j)
                acc[j] += lo[w2][row][cv0 + j] * f;
        }
        const float inv = 1.0f / lstar;
        #pragma unroll
        for (int j = 0; j < 4; ++j)
            out[((size_t)b * CV_N + cv0 + j) * HW_N + q0 + row] = acc[j] * inv;
    }
}

extern "C" void kernel_launch(void* const* d_in, const int* in_sizes, int n_in,
                              void* d_out, int out_size, void* d_ws, size_t ws_size,
                              hipStream_t stream) {
    const float* mem_keys  = (const float*)d_in[0];   // (9,4,64,32,32)
    const float* mem_vals  = (const float*)d_in[1];   // (9,4,64,32,32)
    const float* query     = (const float*)d_in[2];   // (4,64,32,32)
    const float* disparity = (const float*)d_in[3];   // (4,1,32,32)
    const int*   seq_index = (const int*)d_in[4];     // (4,9,2)
    float*       out       = (float*)d_out;           // (4,64,32,32)

    (void)in_sizes; (void)n_in; (void)out_size; (void)d_ws; (void)ws_size;

    dim3 grid(B_N * (HW_N / 16));   // 256 blocks: one 16-query tile each
    dim3 block(32 * NW);            // 8 split-K waves per block -> 2048 waves
    cfa_fp32_wmma_splitk<<<grid, block, 0, stream>>>(
        mem_keys, mem_vals, query, disparity, seq_index, out);
}